// Caps1D_30580167148199
// MI455X (gfx1250) — compile-verified
//
#include <hip/hip_runtime.h>

typedef float v2f __attribute__((ext_vector_type(2)));
typedef float v8f __attribute__((ext_vector_type(8)));

#define B_TOT   1024
#define R_TOT   2336
#define RP      2337      // padded LDS pitch
#define CIN     4
#define COUT    16
#define K_TOT   2
#define MTILE   16
#define NWAVES  8
#define NTHREADS 256
#define NITER   3

// One workgroup: tile of 16 batch rows, one class k.
// u_ji recomputed from L2 via V_WMMA_F32_16X16X4_F32; routing logits in LDS.
// Route loops are 2-way unrolled with dual WMMA accumulator chains for ILP.
__global__ __launch_bounds__(NTHREADS)
void caps_route_kernel(const float* __restrict__ u,   // [B,R,4]
                       const float* __restrict__ W,   // [K,R,4,16]
                       float* __restrict__ cls)       // [B,K]  (||v|| pre-softmax)
{
    __shared__ float blog[MTILE][RP];
    __shared__ float s_lds[MTILE][COUT];
    __shared__ float v_lds[MTILE][COUT];
    __shared__ float rowmax[MTILE];
    __shared__ float invsum[MTILE];

    const int bt   = blockIdx.x;
    const int k    = blockIdx.y;
    const int tid  = threadIdx.x;
    const int wave = tid >> 5;
    const int lane = tid & 31;
    const int half = lane >> 4;
    const int l16  = lane & 15;

    for (int i = tid; i < MTILE * RP; i += NTHREADS)
        ((float*)blog)[i] = 0.0f;
    __syncthreads();

    // A[M=l16][Kd=j+2*half] = u[bt*16+l16][r][2*half + j]
    const float* uBase = u + ((size_t)(bt * MTILE + l16) * R_TOT) * CIN + 2 * half;
    // B[Kd=j+2*half][N=l16] = W[k][r][2*half + j][l16]
    const float* wBase = W + ((size_t)k * R_TOT) * (CIN * COUT);
    const int woff0 = (2 * half + 0) * COUT + l16;
    const int woff1 = (2 * half + 1) * COUT + l16;

    for (int it = 0; it < NITER; ++it) {
        // ---- softmax stats over routes; wave w owns rows m=w, m=w+8 ----
        for (int mi = 0; mi < 2; ++mi) {
            int m = wave + 8 * mi;
            float mx = -3.4e38f;
            for (int r = lane; r < R_TOT; r += 32)
                mx = fmaxf(mx, blog[m][r]);
            for (int off = 16; off > 0; off >>= 1)
                mx = fmaxf(mx, __shfl_xor(mx, off, 32));
            float sm = 0.0f;
            for (int r = lane; r < R_TOT; r += 32)
                sm += __expf(blog[m][r] - mx);
            for (int off = 16; off > 0; off >>= 1)
                sm += __shfl_xor(sm, off, 32);
            if (lane == 0) { rowmax[m] = mx; invsum[m] = 1.0f / sm; }
        }
        for (int i = tid; i < MTILE * COUT; i += NTHREADS)
            ((float*)s_lds)[i] = 0.0f;
        __syncthreads();

        const float rmA = rowmax[l16];
        const float isA = invsum[l16];

        // ---- s = sum_r c[m][r]*u_ji[m][r][:]; c folded into A rows; ----
        // ---- two independent WMMA accumulator chains (routes r, r+8) ----
        v8f sacc0 = {}, sacc1 = {};
        for (int r = wave; r < R_TOT; r += 2 * NWAVES) {
            const int r1 = r + NWAVES;
            v2f a0 = *(const v2f*)(uBase + (size_t)r  * CIN);
            v2f a1 = *(const v2f*)(uBase + (size_t)r1 * CIN);
            const float* wp0 = wBase + (size_t)r  * (CIN * COUT);
            const float* wp1 = wBase + (size_t)r1 * (CIN * COUT);
            v2f bf0, bf1;
            bf0.x = wp0[woff0]; bf0.y = wp0[woff1];
            bf1.x = wp1[woff0]; bf1.y = wp1[woff1];
            float c0 = __expf(blog[l16][r]  - rmA) * isA;
            float c1 = __expf(blog[l16][r1] - rmA) * isA;
            a0.x *= c0; a0.y *= c0;
            a1.x *= c1; a1.y *= c1;
            sacc0 = __builtin_amdgcn_wmma_f32_16x16x4_f32(
                false, a0, false, bf0, (short)0, sacc0, false, false);
            sacc1 = __builtin_amdgcn_wmma_f32_16x16x4_f32(
                false, a1, false, bf1, (short)0, sacc1, false, false);
        }
        #pragma unroll
        for (int j = 0; j < 8; ++j)
            atomicAdd(&s_lds[j + 8 * half][l16], sacc0[j] + sacc1[j]);
        __syncthreads();

        // ---- v = squash(s); final iter: classes = ||v|| = n/(1+n) ----
        if (tid < MTILE) {
            float n = 0.0f;
            for (int o = 0; o < COUT; ++o) { float x = s_lds[tid][o]; n += x * x; }
            float scale = (n / (1.0f + n)) * rsqrtf(fmaxf(n, 1e-30f));
            for (int o = 0; o < COUT; ++o) v_lds[tid][o] = s_lds[tid][o] * scale;
            if (it == NITER - 1)
                cls[(size_t)(bt * MTILE + tid) * K_TOT + k] = n / (1.0f + n);
        }
        __syncthreads();
        if (it == NITER - 1) break;

        // ---- b[m][r] += sum_o u_ji[m][r][o] * v[m][o], 2-way unrolled ----
        for (int r = wave; r < R_TOT; r += 2 * NWAVES) {
            const int r1 = r + NWAVES;
            v2f a0 = *(const v2f*)(uBase + (size_t)r  * CIN);
            v2f a1 = *(const v2f*)(uBase + (size_t)r1 * CIN);
            const float* wp0 = wBase + (size_t)r  * (CIN * COUT);
            const float* wp1 = wBase + (size_t)r1 * (CIN * COUT);
            v2f bf0, bf1;
            bf0.x = wp0[woff0]; bf0.y = wp0[woff1];
            bf1.x = wp1[woff0]; bf1.y = wp1[woff1];
            v8f z0 = {}, z1 = {};
            v8f d0 = __builtin_amdgcn_wmma_f32_16x16x4_f32(
                false, a0, false, bf0, (short)0, z0, false, false);
            v8f d1 = __builtin_amdgcn_wmma_f32_16x16x4_f32(
                false, a1, false, bf1, (short)0, z1, false, false);
            float t0[8], t1[8];
            #pragma unroll
            for (int j = 0; j < 8; ++j) {
                float vv = v_lds[j + 8 * half][l16];
                t0[j] = d0[j] * vv;
                t1[j] = d1[j] * vv;
            }
            #pragma unroll
            for (int mask = 1; mask < 16; mask <<= 1)
                #pragma unroll
                for (int j = 0; j < 8; ++j) {
                    t0[j] += __shfl_xor(t0[j], mask, 32);
                    t1[j] += __shfl_xor(t1[j], mask, 32);
                }
            if (l16 == 0) {
                #pragma unroll
                for (int j = 0; j < 8; ++j) {
                    blog[j + 8 * half][r]  += t0[j];
                    blog[j + 8 * half][r1] += t1[j];
                }
            }
        }
        __syncthreads();
    }
}

// In-place softmax over the 2 classes, directly on d_out.
__global__ void caps_softmax_kernel(float* __restrict__ out)
{
    int b = blockIdx.x * blockDim.x + threadIdx.x;
    if (b >= B_TOT) return;
    float c0 = out[b * 2 + 0], c1 = out[b * 2 + 1];
    float mx = fmaxf(c0, c1);
    float e0 = __expf(c0 - mx), e1 = __expf(c1 - mx);
    float inv = 1.0f / (e0 + e1);
    out[b * 2 + 0] = e0 * inv;
    out[b * 2 + 1] = e1 * inv;
}

extern "C" void kernel_launch(void* const* d_in, const int* in_sizes, int n_in,
                              void* d_out, int out_size, void* d_ws, size_t ws_size,
                              hipStream_t stream) {
    (void)in_sizes; (void)n_in; (void)d_ws; (void)ws_size; (void)out_size;
    const float* u = (const float*)d_in[0];   // [1024,2336,4]
    const float* W = (const float*)d_in[1];   // [2,2336,4,16]
    float* out = (float*)d_out;               // [1024,2]

    dim3 grid(B_TOT / MTILE, K_TOT);          // 64 x 2 workgroups
    caps_route_kernel<<<grid, NTHREADS, 0, stream>>>(u, W, out);
    caps_softmax_kernel<<<(B_TOT + 255) / 256, 256, 0, stream>>>(out);
}